// PhiQComputer_41592463294696
// MI455X (gfx1250) — compile-verified
//
#include <hip/hip_runtime.h>
#include <math.h>

typedef __attribute__((ext_vector_type(2))) float v2f;
typedef __attribute__((ext_vector_type(8))) float v8f;

#define NSITES 16
#define KDIM   256
#define WPB    8        // waves (= batch elements) per block
#define EPSF   1e-10f

// Fully-unrolled cyclic Jacobi eigen-solve on an NxN symmetric matrix held in
// registers (all indices compile-time constants after unrolling; only the
// sweep loop is runtime). Returns the Shannon entropy of the normalized
// eigenvalue distribution, matching the reference's (s^2+EPS) handling.
template <int N>
__device__ __forceinline__ float jacobi_entropy(float (&A)[N][N], int sweeps) {
  for (int sw = 0; sw < sweeps; ++sw) {
#pragma unroll
    for (int p = 0; p < N - 1; ++p) {
#pragma unroll
      for (int q = p + 1; q < N; ++q) {
        float apq = A[p][q];
        float c = 1.0f, s = 0.0f;
        if (__builtin_fabsf(apq) > 1e-30f) {
          float tau = (A[q][q] - A[p][p]) / (2.0f * apq);
          float t = __builtin_copysignf(1.0f, tau) /
                    (__builtin_fabsf(tau) + __builtin_sqrtf(tau * tau + 1.0f));
          c = 1.0f / __builtin_sqrtf(t * t + 1.0f);
          s = t * c;
        }
        // A <- J^T A : rotate rows p,q
#pragma unroll
        for (int k = 0; k < N; ++k) {
          float x = A[p][k], y = A[q][k];
          A[p][k] = c * x - s * y;
          A[q][k] = s * x + c * y;
        }
        // A <- A J : rotate cols p,q
#pragma unroll
        for (int k = 0; k < N; ++k) {
          float x = A[k][p], y = A[k][q];
          A[k][p] = c * x - s * y;
          A[k][q] = s * x + c * y;
        }
        A[p][q] = 0.0f;
        A[q][p] = 0.0f;
      }
    }
  }
  float lam[N];
  float sum = 0.0f;
#pragma unroll
  for (int i = 0; i < N; ++i) {
    float l = A[i][i];
    l = l > 0.0f ? l : 0.0f;   // eigenvalues of PSD Gram; clamp rounding noise
    lam[i] = l + EPSF;          // s^2 + EPS
    sum += lam[i];
  }
  float inv = 1.0f / sum;
  float S = 0.0f;
#pragma unroll
  for (int i = 0; i < N; ++i) {
    float pr = lam[i] * inv;
    S -= pr * __builtin_logf(pr + EPSF);
  }
  return S;
}

// G = C * C^T for an NxN block stored row-major in LDS.
template <int N>
__device__ __forceinline__ void gram(const float* __restrict__ s, float (&G)[N][N]) {
  float C[N][N];
#pragma unroll
  for (int i = 0; i < N; ++i)
#pragma unroll
    for (int j = 0; j < N; ++j) C[i][j] = s[i * N + j];
#pragma unroll
  for (int i = 0; i < N; ++i)
#pragma unroll
    for (int j = 0; j <= i; ++j) {
      float d = 0.0f;
#pragma unroll
      for (int k = 0; k < N; ++k) d += C[i][k] * C[j][k];
      G[i][j] = d;
      G[j][i] = d;
    }
}

__global__ __launch_bounds__(256) void phiq_kernel(const float* __restrict__ sites,
                                                   float* __restrict__ out) {
  // Per block: 8 waves -> 8 batch elements. LDS slot: 96 floats per element:
  //   [0..63]  corr_whole = S[0:8, 8:16]   (row-major 8x8)
  //   [64..79] corr_left  = S[0:4, 4:8]    (row-major 4x4)
  //   [80..95] corr_right = S[8:12,12:16]  (row-major 4x4)
  __shared__ float smat[WPB][96];

  const int tid  = threadIdx.x;
  const int w    = tid >> 5;
  const int lane = tid & 31;
  const int b    = blockIdx.x * WPB + w;
  const int row  = lane & 15;          // M (and N) index this lane serves
  const int ko   = (lane >> 4) << 1;   // K sub-offset: 0 for lanes 0-15, 2 for 16-31

  // --- Phase 1: S = sites * sites^T (16x16, K=256) via f32 WMMA ---
  // A fragment (16x4 f32): lane L, VGPR j = sites[L&15][4*kc + 2*(L>=16) + j].
  // For a Gram product the matching B fragment (4x16) has identical contents,
  // so one float2 load per lane per K-step feeds both operands.
  const float* base = sites + (size_t)b * (NSITES * KDIM) + (size_t)row * KDIM + ko;

  v8f acc = {};
  // 16-wide unroll: one 16-load s_clause feeding 16 back-to-back WMMAs per
  // trip halves the per-wave exposed-latency fraction vs unroll-8, at a VGPR
  // cost (+32) that stays under the Jacobi phase's register ceiling.
#pragma unroll 16
  for (int kc = 0; kc < KDIM / 4; ++kc) {
    v2f a = *(const v2f*)(base + kc * 4);
    acc = __builtin_amdgcn_wmma_f32_16x16x4_f32(
        /*neg_a=*/false, a, /*neg_b=*/false, a,
        /*c_mod=*/(short)0, acc, /*reuse_a=*/false, /*reuse_b=*/false);
  }

  // Accumulator layout: VGPR v holds S[M][N] with M = v + 8*(lane>=16),
  // N = lane&15. Extract the three corr submatrices into LDS.
  float* s = smat[w];
  if (lane < 16) {
    if (row >= 8) {           // corr_whole rows 0..7, col = row-8
#pragma unroll
      for (int v = 0; v < 8; ++v) s[v * 8 + (row - 8)] = acc[v];
    } else if (row >= 4) {    // corr_left rows 0..3, col = row-4
#pragma unroll
      for (int v = 0; v < 4; ++v) s[64 + v * 4 + (row - 4)] = acc[v];
    }
  } else {
    if (row >= 12) {          // corr_right rows 8..11 (v=0..3), col = row-12
#pragma unroll
      for (int v = 0; v < 4; ++v) s[80 + v * 4 + (row - 12)] = acc[v];
    }
  }
  __syncthreads();

  // --- Phase 2: one thread per batch element: eigenvalues + entropies ---
  if (tid < WPB) {
    const float* sm = smat[tid];

    float A8[8][8];
    gram<8>(sm, A8);
    float Sw = jacobi_entropy<8>(A8, 8);

    float A4[4][4];
    gram<4>(sm + 64, A4);
    float Sl = jacobi_entropy<4>(A4, 6);

    gram<4>(sm + 80, A4);
    float Sr = jacobi_entropy<4>(A4, 6);

    float phi = Sw - (Sl + Sr);
    out[blockIdx.x * WPB + tid] = phi > 0.0f ? phi : 0.0f;
  }
}

extern "C" void kernel_launch(void* const* d_in, const int* in_sizes, int n_in,
                              void* d_out, int out_size, void* d_ws, size_t ws_size,
                              hipStream_t stream) {
  (void)n_in; (void)d_ws; (void)ws_size; (void)out_size;
  const float* sites = (const float*)d_in[0];
  float* out = (float*)d_out;
  const int batch = in_sizes[0] / (NSITES * KDIM);   // 8192
  const int blocks = batch / WPB;                    // 1024 blocks x 256 threads
  phiq_kernel<<<blocks, 256, 0, stream>>>(sites, out);
}